// Ewald_46969762349287
// MI455X (gfx1250) — compile-verified
//
#include <hip/hip_runtime.h>

typedef float v2f __attribute__((ext_vector_type(2)));
typedef float v8f __attribute__((ext_vector_type(8)));

#define PI_F        3.14159265358979323846f
#define TWO_PI_F    6.28318530717958647692f
#define ALPHA_F     0.3f
#define INV_SQRTPI  0.56418958354775628695f

// ---------------------------------------------------------------- helpers ---
static __device__ __forceinline__ v8f wmma4(v2f a, v2f b, v8f c) {
  // V_WMMA_F32_16X16X4_F32 : D = A(16x4) * B(4x16) + C(16x16)
  return __builtin_amdgcn_wmma_f32_16x16x4_f32(false, a, false, b, (short)0, c,
                                               false, false);
}

static __device__ __forceinline__ v8f splat8(float x) {
  v8f r;
#pragma unroll
  for (int i = 0; i < 8; ++i) r[i] = x;
  return r;
}

static __device__ __forceinline__ float det3x3(const float* b) {
  return b[0] * (b[4] * b[8] - b[5] * b[7])
       - b[1] * (b[3] * b[8] - b[5] * b[6])
       + b[2] * (b[3] * b[7] - b[4] * b[6]);
}

static __device__ __forceinline__ void inv3x3(const float* b, float inv[9]) {
  float id = 1.0f / det3x3(b);
  inv[0] =  (b[4] * b[8] - b[5] * b[7]) * id;
  inv[1] = -(b[1] * b[8] - b[2] * b[7]) * id;
  inv[2] =  (b[1] * b[5] - b[2] * b[4]) * id;
  inv[3] = -(b[3] * b[8] - b[5] * b[6]) * id;
  inv[4] =  (b[0] * b[8] - b[2] * b[6]) * id;
  inv[5] = -(b[0] * b[5] - b[2] * b[3]) * id;
  inv[6] =  (b[3] * b[7] - b[4] * b[6]) * id;
  inv[7] = -(b[0] * b[7] - b[1] * b[6]) * id;
  inv[8] =  (b[0] * b[4] - b[1] * b[3]) * id;
}

// ------------------------------------------------------------------ pass 1 ---
// One block per 16-k tile. 8 waves split the atom tiles; each wave accumulates
// S_real/S_imag contributions for its 16 k-rows via 4 f32 WMMAs per atom tile.
__global__ __launch_bounds__(256) void ewald_pass1(
    const float* __restrict__ coords, const float* __restrict__ box,
    const float* __restrict__ q, const float* __restrict__ p,
    const float* __restrict__ t, const float* __restrict__ kvecs,
    float* __restrict__ kvt, float* __restrict__ Srg, float* __restrict__ Sig,
    float* __restrict__ ek, int N, int K, int ntA) {
  const int lane = threadIdx.x & 31;
  const int wave = threadIdx.x >> 5;
  const int half = lane >> 4;     // 0: lanes 0-15, 1: lanes 16-31
  const int m    = lane & 15;     // row (k) within tile / column (atom) index
  const int ktile = blockIdx.x;
  const int krow  = ktile * 16 + m;

  float binv[9];
  inv3x3(box, binv);

  // transformed k-vector for this lane's row (zero for padded rows)
  float kx = 0.f, ky = 0.f, kz = 0.f;
  if (krow < K) {
    float a0 = kvecs[3 * krow], a1 = kvecs[3 * krow + 1], a2 = kvecs[3 * krow + 2];
    kx = a0 * binv[0] + a1 * binv[3] + a2 * binv[6];
    ky = a0 * binv[1] + a1 * binv[4] + a2 * binv[7];
    kz = a0 * binv[2] + a1 * binv[5] + a2 * binv[8];
  }

  // A matrices (fixed for the block): K0,K1 in lanes<16 ; K2,K3 in lanes>=16
  const float cq = -(TWO_PI_F * TWO_PI_F) / 3.0f;
  v2f a_ph, a_li, a_q1, a_q2;
  a_ph.x = half ? kz : kx;                 a_ph.y = half ? 0.f : ky;
  a_li.x = TWO_PI_F * (half ? kz : kx);    a_li.y = half ? 0.f : TWO_PI_F * ky;
  a_q1.x = cq * (half ? kz * kz : kx * kx);
  a_q1.y = cq * (half ? kx * ky : ky * ky);
  a_q2.x = half ? 0.f : cq * kx * kz;
  a_q2.y = half ? 0.f : cq * ky * kz;

  float er[8], ei[8];
#pragma unroll
  for (int j = 0; j < 8; ++j) { er[j] = 0.f; ei[j] = 0.f; }

  for (int tile = wave; tile < ntA; tile += 8) {
    int n = tile * 16 + m;
    int nc = (n < N) ? n : 0;
    float msk = (n < N) ? 1.0f : 0.0f;

    float x = coords[3 * nc], y = coords[3 * nc + 1], z = coords[3 * nc + 2];
    float qa = q[nc] * msk;
    float px = p[3 * nc] * msk, py = p[3 * nc + 1] * msk, pz = p[3 * nc + 2] * msk;
    const float* tn = t + 9 * nc;
    float txx = tn[0] * msk, tyy = tn[4] * msk, tzz = tn[8] * msk;
    float sxy = (tn[1] + tn[3]) * msk;
    float sxz = (tn[2] + tn[6]) * msk;
    float syz = (tn[5] + tn[7]) * msk;

    v2f b_ph, b_li, b_q1, b_q2;
    b_ph.x = half ? z : x;     b_ph.y = half ? 0.f : y;
    b_li.x = half ? pz : px;   b_li.y = half ? 0.f : py;
    b_q1.x = half ? tzz : txx; b_q1.y = half ? sxy : tyy;
    b_q2.x = half ? 0.f : sxz; b_q2.y = half ? 0.f : syz;

    v8f d_ph = wmma4(a_ph, b_ph, splat8(0.f));       // k . r
    v8f d_li = wmma4(a_li, b_li, splat8(0.f));       // 2pi k . p
    v8f d_lr = wmma4(a_q1, b_q1, splat8(qa));        // q - (4pi^2/3) k^T t k
    d_lr     = wmma4(a_q2, b_q2, d_lr);

#pragma unroll
    for (int j = 0; j < 8; ++j) {
      float ph = __builtin_amdgcn_fractf(d_ph[j]);   // v_sin/cos take revolutions
      float c  = __builtin_amdgcn_cosf(ph);
      float s  = __builtin_amdgcn_sinf(ph);
      er[j] = __builtin_fmaf(c, d_lr[j], er[j]);
      er[j] = __builtin_fmaf(-s, d_li[j], er[j]);
      ei[j] = __builtin_fmaf(c, d_li[j], ei[j]);
      ei[j] = __builtin_fmaf(s, d_lr[j], ei[j]);
    }
  }

  // reduce over the 16 columns held in each half-wave
#pragma unroll
  for (int j = 0; j < 8; ++j) {
    float r = er[j], im = ei[j];
#pragma unroll
    for (int off = 1; off < 16; off <<= 1) {
      r  += __shfl_xor(r, off, 32);
      im += __shfl_xor(im, off, 32);
    }
    er[j] = r; ei[j] = im;
  }

  __shared__ float red_r[8][16];
  __shared__ float red_i[8][16];
  if (lane == 0) {
#pragma unroll
    for (int j = 0; j < 8; ++j) { red_r[wave][j] = er[j]; red_i[wave][j] = ei[j]; }
  }
  if (lane == 16) {
#pragma unroll
    for (int j = 0; j < 8; ++j) { red_r[wave][8 + j] = er[j]; red_i[wave][8 + j] = ei[j]; }
  }
  __syncthreads();

  if (threadIdx.x < 16) {   // these threads are wave 0, half 0 -> kx,ky,kz valid
    int kk = ktile * 16 + threadIdx.x;
    float Sr = 0.f, Si = 0.f;
#pragma unroll
    for (int w = 0; w < 8; ++w) { Sr += red_r[w][threadIdx.x]; Si += red_i[w][threadIdx.x]; }
    float g = 0.f, e = 0.f, srg = 0.f, sig = 0.f;
    if (kk < K) {
      float ksq = kx * kx + ky * ky + kz * kz;
      g = __expf(-(PI_F * PI_F) * ksq / (ALPHA_F * ALPHA_F)) / ksq;
      srg = g * Sr; sig = g * Si; e = g * (Sr * Sr + Si * Si);
    }
    Srg[kk] = srg; Sig[kk] = sig; ek[kk] = e;
    kvt[3 * kk] = kx; kvt[3 * kk + 1] = ky; kvt[3 * kk + 2] = kz;  // 0 on pad
  }
}

// ------------------------------------------------------------------ pass 2 ---
// One block per 16-atom tile; atoms are WMMA rows, k are columns so the per-k
// data (Srg/Sig/kvt) loads coalesced per lane. 8 waves split the k tiles.
__global__ __launch_bounds__(256) void ewald_pass2(
    const float* __restrict__ coords, const float* __restrict__ box,
    const float* __restrict__ q, const float* __restrict__ p,
    const float* __restrict__ kvt, const float* __restrict__ Srg,
    const float* __restrict__ Sig, float* __restrict__ out, int N, int ntK) {
  const int lane = threadIdx.x & 31;
  const int wave = threadIdx.x >> 5;
  const int half = lane >> 4;
  const int m    = lane & 15;
  const int atile = blockIdx.x;
  const int arow  = atile * 16 + m;
  const int ac    = (arow < N) ? arow : 0;

  float x = coords[3 * ac], y = coords[3 * ac + 1], z = coords[3 * ac + 2];
  v2f a_ph;
  a_ph.x = half ? z : x;
  a_ph.y = half ? 0.f : y;

  float pot[8], fx[8], fy[8], fz[8];
#pragma unroll
  for (int j = 0; j < 8; ++j) { pot[j] = 0.f; fx[j] = 0.f; fy[j] = 0.f; fz[j] = 0.f; }

  for (int kt = wave; kt < ntK; kt += 8) {
    int kk = kt * 16 + m;                       // this lane's column k
    float kvx = kvt[3 * kk], kvy = kvt[3 * kk + 1], kvz = kvt[3 * kk + 2];
    float sr = Srg[kk], si = Sig[kk];           // zero on padded k -> no effect

    v2f b;
    b.x = half ? kvz : kvx;
    b.y = half ? 0.f : kvy;
    v8f d_ph = wmma4(a_ph, b, splat8(0.f));     // k . r tile (atoms x k)

#pragma unroll
    for (int j = 0; j < 8; ++j) {
      float ph = __builtin_amdgcn_fractf(d_ph[j]);
      float c  = __builtin_amdgcn_cosf(ph);
      float s  = __builtin_amdgcn_sinf(ph);
      pot[j] = __builtin_fmaf(sr, c, pot[j]);
      pot[j] = __builtin_fmaf(si, s, pot[j]);
      float kim = sr * s;
      kim = __builtin_fmaf(-si, c, kim);        // Sr*sin - Si*cos
      fx[j] = __builtin_fmaf(kim, kvx, fx[j]);
      fy[j] = __builtin_fmaf(kim, kvy, fy[j]);
      fz[j] = __builtin_fmaf(kim, kvz, fz[j]);
    }
  }

#pragma unroll
  for (int j = 0; j < 8; ++j) {
    float a = pot[j], b0 = fx[j], c0 = fy[j], d0 = fz[j];
#pragma unroll
    for (int off = 1; off < 16; off <<= 1) {
      a  += __shfl_xor(a, off, 32);
      b0 += __shfl_xor(b0, off, 32);
      c0 += __shfl_xor(c0, off, 32);
      d0 += __shfl_xor(d0, off, 32);
    }
    pot[j] = a; fx[j] = b0; fy[j] = c0; fz[j] = d0;
  }

  __shared__ float rp[8][16], rx[8][16], ry[8][16], rz[8][16];
  if (lane == 0) {
#pragma unroll
    for (int j = 0; j < 8; ++j) {
      rp[wave][j] = pot[j]; rx[wave][j] = fx[j]; ry[wave][j] = fy[j]; rz[wave][j] = fz[j];
    }
  }
  if (lane == 16) {
#pragma unroll
    for (int j = 0; j < 8; ++j) {
      rp[wave][8 + j] = pot[j]; rx[wave][8 + j] = fx[j];
      ry[wave][8 + j] = fy[j];  rz[wave][8 + j] = fz[j];
    }
  }
  __syncthreads();

  if (threadIdx.x < 16) {
    int atom = atile * 16 + threadIdx.x;
    if (atom < N) {
      float sp = 0.f, sx = 0.f, sy = 0.f, sz = 0.f;
#pragma unroll
      for (int w = 0; w < 8; ++w) {
        sp += rp[w][threadIdx.x]; sx += rx[w][threadIdx.x];
        sy += ry[w][threadIdx.x]; sz += rz[w][threadIdx.x];
      }
      float V = det3x3(box);
      float a_rpi = ALPHA_F * INV_SQRTPI;
      float cpot = 2.0f / (PI_F * V);
      float cfld = 4.0f / V;
      float cselfp = a_rpi * (4.0f * ALPHA_F * ALPHA_F / 3.0f);
      out[1 + atom] = cpot * sp - 2.0f * a_rpi * q[atom];
      float* f = out + 1 + N + 3 * atom;
      f[0] = cfld * sx + cselfp * p[3 * atom];
      f[1] = cfld * sy + cselfp * p[3 * atom + 1];
      f[2] = cfld * sz + cselfp * p[3 * atom + 2];
    }
  }
}

// ------------------------------------------------------------------ energy ---
__global__ __launch_bounds__(256) void ewald_energy(
    const float* __restrict__ q, const float* __restrict__ p,
    const float* __restrict__ t, const float* __restrict__ box,
    const float* __restrict__ ek, float* __restrict__ out, int N, int K) {
  const int tid = threadIdx.x;
  float acc = 0.f;
  for (int i = tid; i < K; i += 256) acc += ek[i];
  float sq = 0.f;
  for (int i = tid; i < N; i += 256) sq += q[i] * q[i];
  float sp = 0.f;
  for (int i = tid; i < 3 * N; i += 256) sp += p[i] * p[i];
  float st = 0.f;
  for (int i = tid; i < 9 * N; i += 256) st += t[i] * t[i];

  float V = det3x3(box);
  float a_rpi = ALPHA_F * INV_SQRTPI;
  float a2 = ALPHA_F * ALPHA_F;
  float local = acc / (PI_F * V)
              - a_rpi * (sq + (2.0f * a2 / 3.0f) * sp
                            + (8.0f * a2 * a2 / 45.0f) * st);

  __shared__ float buf[256];
  buf[tid] = local;
  __syncthreads();
  for (int s = 128; s > 0; s >>= 1) {
    if (tid < s) buf[tid] += buf[tid + s];
    __syncthreads();
  }
  if (tid == 0) out[0] = buf[0];
}

// ------------------------------------------------------------------ launch ---
extern "C" void kernel_launch(void* const* d_in, const int* in_sizes, int n_in,
                              void* d_out, int out_size, void* d_ws, size_t ws_size,
                              hipStream_t stream) {
  const float* coords = (const float*)d_in[0];
  const float* box    = (const float*)d_in[1];
  const float* q      = (const float*)d_in[2];
  const float* p      = (const float*)d_in[3];
  const float* t      = (const float*)d_in[4];
  const float* kvecs  = (const float*)d_in[5];
  float* out = (float*)d_out;

  int N = in_sizes[2];          // atoms
  int K = in_sizes[5] / 3;      // k-vectors
  int ntA = (N + 15) / 16;      // atom tiles
  int ntK = (K + 15) / 16;      // k tiles
  int KP  = ntK * 16;           // padded K

  float* ws  = (float*)d_ws;
  float* kvt = ws;              // KP * 3
  float* Srg = kvt + 3 * KP;    // KP  (gaussian * S_real)
  float* Sig = Srg + KP;        // KP  (gaussian * S_imag)
  float* ek  = Sig + KP;        // KP  (gaussian * |S|^2)

  ewald_pass1<<<ntK, 256, 0, stream>>>(coords, box, q, p, t, kvecs,
                                       kvt, Srg, Sig, ek, N, K, ntA);
  ewald_energy<<<1, 256, 0, stream>>>(q, p, t, box, ek, out, N, K);
  ewald_pass2<<<ntA, 256, 0, stream>>>(coords, box, q, p, kvt, Srg, Sig,
                                       out, N, ntK);
}